// SplineCNN_43654047596705
// MI455X (gfx1250) — compile-verified
//
#include <hip/hip_runtime.h>
#include <math.h>

#define NN 100000
#define NE 1600000
#define FIN 64
#define HID 64
#define CLS 40

typedef __attribute__((ext_vector_type(2))) float v2f;
typedef __attribute__((ext_vector_type(8))) float v8f;

// ---- workspace layout (in floats) ----
// hcat1: NN*192  (h0|h1|xroot for layer1; reused as gcat stride 144 for layer2)
// agg1 : NN*64
// deg  : NN
// agg2 : NN*40
// hact : NN*64
#define OFF_HCAT 0ull
#define OFF_AGG1 (OFF_HCAT + (unsigned long long)NN * 192ull)
#define OFF_DEG  (OFF_AGG1 + (unsigned long long)NN * 64ull)
#define OFF_AGG2 (OFF_DEG  + (unsigned long long)NN)
#define OFF_HACT (OFF_AGG2 + (unsigned long long)NN * 40ull)
#define ZERO_LEN ((unsigned long long)NN * 64ull + (unsigned long long)NN + (unsigned long long)NN * 40ull)

__device__ __forceinline__ void atomAddF(float* p, float v) {
  unsafeAtomicAdd(p, v);   // lowers to global_atomic_add_f32 on gfx1250
}

// -------- zero scratch (agg1 | deg | agg2 are contiguous) --------
__global__ void zero_kernel(float* __restrict__ p, unsigned long long n) {
  unsigned long long i = (unsigned long long)blockIdx.x * blockDim.x + threadIdx.x;
  if (i < n) p[i] = 0.0f;
}

// -------- layer-1 dense transform: hcat[n, 0:64]=x@W1[0], [64:128]=x@W1[1], [128:192]=x@root1
// one wave per (16-row tile, 16-col tile job); 12 jobs = 3 matrices x 4 col tiles
__global__ __launch_bounds__(32) void gemm1_kernel(const float* __restrict__ x,
                                                   const float* __restrict__ W1,
                                                   const float* __restrict__ root1,
                                                   float* __restrict__ hcat) {
  const int rt   = blockIdx.x;
  const int job  = blockIdx.y;        // 0..11
  const int mat  = job >> 2;          // 0,1 -> W1[k]; 2 -> root1
  const int lct  = job & 3;           // local 16-col tile
  const int lane = threadIdx.x;
  const int mrow = lane & 15;         // A row / B col within tile
  const int kb   = (lane >> 4) * 2;   // K sub-offset per lane half
  const float* W = (mat < 2) ? (W1 + (size_t)mat * FIN * HID) : root1;
  const int col  = lct * 16 + mrow;
  const float* arow = x + ((size_t)rt * 16 + mrow) * FIN;

  v8f acc = {0.f, 0.f, 0.f, 0.f, 0.f, 0.f, 0.f, 0.f};
  for (int k = 0; k < FIN; k += 4) {
    v2f a, b;
    a.x = arow[k + kb];
    a.y = arow[k + kb + 1];
    b.x = W[(size_t)(k + kb) * HID + col];
    b.y = W[(size_t)(k + kb + 1) * HID + col];
    acc = __builtin_amdgcn_wmma_f32_16x16x4_f32(false, a, false, b, (short)0, acc,
                                                false, false);
  }
  const int outcol = mat * 64 + col;
  const int rbase  = rt * 16 + (lane >> 4) * 8;
#pragma unroll
  for (int j = 0; j < 8; ++j)
    hcat[(size_t)(rbase + j) * 192 + outcol] = acc[j];
}

// -------- layer-2 dense transform: gcat[n, 0:48|48:96|96:144] = hact@{W2[0],W2[1],root2} (40 valid cols)
__global__ __launch_bounds__(32) void gemm2_kernel(const float* __restrict__ hact,
                                                   const float* __restrict__ W2,
                                                   const float* __restrict__ root2,
                                                   float* __restrict__ gcat) {
  const int rt   = blockIdx.x;
  const int job  = blockIdx.y;        // 0..8
  const int mat  = job / 3;
  const int lct  = job % 3;
  const int lane = threadIdx.x;
  const int mrow = lane & 15;
  const int kb   = (lane >> 4) * 2;
  const float* W = (mat < 2) ? (W2 + (size_t)mat * HID * CLS) : root2;
  const int col  = lct * 16 + mrow;
  const float msk = (col < CLS) ? 1.0f : 0.0f;  // value-select: EXEC stays full for WMMA
  const int  ccol = (col < CLS) ? col : (CLS - 1);
  const float* arow = hact + ((size_t)rt * 16 + mrow) * HID;

  v8f acc = {0.f, 0.f, 0.f, 0.f, 0.f, 0.f, 0.f, 0.f};
  for (int k = 0; k < HID; k += 4) {
    v2f a, b;
    a.x = arow[k + kb];
    a.y = arow[k + kb + 1];
    b.x = W[(size_t)(k + kb) * CLS + ccol] * msk;
    b.y = W[(size_t)(k + kb + 1) * CLS + ccol] * msk;
    acc = __builtin_amdgcn_wmma_f32_16x16x4_f32(false, a, false, b, (short)0, acc,
                                                false, false);
  }
  if (col < CLS) {
    const int outcol = mat * 48 + col;
    const int rbase  = rt * 16 + (lane >> 4) * 8;
#pragma unroll
    for (int j = 0; j < 8; ++j)
      gcat[(size_t)(rbase + j) * 144 + outcol] = acc[j];
  }
}

// -------- layer-1 edge scatter: one wave per edge, 2 channels/lane --------
__global__ __launch_bounds__(256) void edge1_kernel(const int* __restrict__ src,
                                                    const int* __restrict__ dst,
                                                    const float* __restrict__ u,
                                                    const float* __restrict__ hcat,
                                                    float* __restrict__ agg,
                                                    float* __restrict__ deg) {
  const long long wid = ((long long)blockIdx.x * blockDim.x + threadIdx.x) >> 5;
  const int lane = threadIdx.x & 31;
  if (wid >= NE) return;
  const int s = src[wid], d = dst[wid];
  const float uu = u[wid], w0 = 1.0f - uu;
  const int c = lane * 2;
  const float* hp = hcat + (size_t)s * 192;
  const float2 h0 = *(const float2*)(hp + c);
  const float2 h1 = *(const float2*)(hp + 64 + c);
  float* ap = agg + (size_t)d * 64 + c;
  atomAddF(ap + 0, w0 * h0.x + uu * h1.x);
  atomAddF(ap + 1, w0 * h0.y + uu * h1.y);
  if (lane == 0) atomAddF(deg + d, 1.0f);
}

// -------- layer-1 finalize: mean + root + bias + ELU --------
__global__ __launch_bounds__(256) void fin1_kernel(const float* __restrict__ agg,
                                                   const float* __restrict__ deg,
                                                   const float* __restrict__ hcat,
                                                   const float* __restrict__ b1,
                                                   float* __restrict__ hact) {
  const long long n = ((long long)blockIdx.x * blockDim.x + threadIdx.x) >> 5;
  const int lane = threadIdx.x & 31;
  if (n >= NN) return;
  const float inv = 1.0f / fmaxf(deg[n], 1.0f);
  const int c = lane * 2;
  const float2 ag = *(const float2*)(agg + (size_t)n * 64 + c);
  const float2 xr = *(const float2*)(hcat + (size_t)n * 192 + 128 + c);
  float v0 = ag.x * inv + xr.x + b1[c];
  float v1 = ag.y * inv + xr.y + b1[c + 1];
  v0 = (v0 > 0.0f) ? v0 : expm1f(v0);     // ELU(alpha=1)
  v1 = (v1 > 0.0f) ? v1 : expm1f(v1);
  float* op = hact + (size_t)n * 64 + c;
  op[0] = v0;
  op[1] = v1;
}

// -------- layer-2 edge scatter (degree reused from layer 1) --------
__global__ __launch_bounds__(256) void edge2_kernel(const int* __restrict__ src,
                                                    const int* __restrict__ dst,
                                                    const float* __restrict__ u,
                                                    const float* __restrict__ gcat,
                                                    float* __restrict__ agg) {
  const long long wid = ((long long)blockIdx.x * blockDim.x + threadIdx.x) >> 5;
  const int lane = threadIdx.x & 31;
  if (wid >= NE) return;
  const int c = lane * 2;
  if (c >= CLS) return;
  const int s = src[wid], d = dst[wid];
  const float uu = u[wid], w0 = 1.0f - uu;
  const float* gp = gcat + (size_t)s * 144;
  const float2 g0 = *(const float2*)(gp + c);
  const float2 g1 = *(const float2*)(gp + 48 + c);
  float* ap = agg + (size_t)d * CLS + c;
  atomAddF(ap + 0, w0 * g0.x + uu * g1.x);
  atomAddF(ap + 1, w0 * g0.y + uu * g1.y);
}

// -------- layer-2 finalize: mean + root + bias, then log_softmax over 40 --------
__global__ __launch_bounds__(256) void fin2_kernel(const float* __restrict__ agg,
                                                   const float* __restrict__ deg,
                                                   const float* __restrict__ gcat,
                                                   const float* __restrict__ b2,
                                                   float* __restrict__ out) {
  const long long n = ((long long)blockIdx.x * blockDim.x + threadIdx.x) >> 5;
  const int lane = threadIdx.x & 31;
  if (n >= NN) return;
  const float inv = 1.0f / fmaxf(deg[n], 1.0f);
  const int c = lane * 2;
  const bool valid = (c < CLS);
  float a0 = -INFINITY, a1 = -INFINITY;
  if (valid) {
    a0 = agg[(size_t)n * CLS + c]     * inv + gcat[(size_t)n * 144 + 96 + c]     + b2[c];
    a1 = agg[(size_t)n * CLS + c + 1] * inv + gcat[(size_t)n * 144 + 96 + c + 1] + b2[c + 1];
  }
  float m = fmaxf(a0, a1);
#pragma unroll
  for (int off = 16; off > 0; off >>= 1) m = fmaxf(m, __shfl_xor(m, off, 32));
  float e = valid ? (expf(a0 - m) + expf(a1 - m)) : 0.0f;
#pragma unroll
  for (int off = 16; off > 0; off >>= 1) e += __shfl_xor(e, off, 32);
  const float L = m + logf(e);
  if (valid) {
    float* op = out + (size_t)n * CLS + c;
    op[0] = a0 - L;
    op[1] = a1 - L;
  }
}

extern "C" void kernel_launch(void* const* d_in, const int* in_sizes, int n_in,
                              void* d_out, int out_size, void* d_ws, size_t ws_size,
                              hipStream_t stream) {
  const float* x     = (const float*)d_in[0];
  const int*   ei    = (const int*)  d_in[1];   // (2, NE): src then dst
  const float* ea    = (const float*)d_in[2];   // (NE, 1)
  const float* W1    = (const float*)d_in[3];
  const float* root1 = (const float*)d_in[4];
  const float* b1    = (const float*)d_in[5];
  const float* W2    = (const float*)d_in[6];
  const float* root2 = (const float*)d_in[7];
  const float* b2    = (const float*)d_in[8];
  float* out = (float*)d_out;
  float* ws  = (float*)d_ws;

  float* hcat = ws + OFF_HCAT;   // layer1: stride 192; reused as layer2 gcat stride 144
  float* agg1 = ws + OFF_AGG1;
  float* deg  = ws + OFF_DEG;
  float* agg2 = ws + OFF_AGG2;
  float* hact = ws + OFF_HACT;
  const int* srcp = ei;
  const int* dstp = ei + NE;

  // 1. zero accumulators + degree (contiguous region)
  {
    unsigned long long n = ZERO_LEN;
    unsigned int blocks = (unsigned int)((n + 255ull) / 256ull);
    zero_kernel<<<blocks, 256, 0, stream>>>(agg1, n);
  }
  // 2. layer-1 dense transforms (WMMA)
  {
    dim3 g(NN / 16, 12);
    gemm1_kernel<<<g, 32, 0, stream>>>(x, W1, root1, hcat);
  }
  // 3. layer-1 edge gather/scatter + degree
  {
    long long waves = NE;
    unsigned int blocks = (unsigned int)((waves * 32 + 255) / 256);
    edge1_kernel<<<blocks, 256, 0, stream>>>(srcp, dstp, ea, hcat, agg1, deg);
  }
  // 4. layer-1 finalize (mean + root + bias + ELU)
  {
    unsigned int blocks = (unsigned int)(((long long)NN * 32 + 255) / 256);
    fin1_kernel<<<blocks, 256, 0, stream>>>(agg1, deg, hcat, b1, hact);
  }
  // 5. layer-2 dense transforms (WMMA), writes into reused hcat buffer
  {
    dim3 g(NN / 16, 9);
    gemm2_kernel<<<g, 32, 0, stream>>>(hact, W2, root2, hcat);
  }
  // 6. layer-2 edge gather/scatter (degree reused)
  {
    long long waves = NE;
    unsigned int blocks = (unsigned int)((waves * 32 + 255) / 256);
    edge2_kernel<<<blocks, 256, 0, stream>>>(srcp, dstp, ea, hcat, agg2);
  }
  // 7. layer-2 finalize + log_softmax
  {
    unsigned int blocks = (unsigned int)(((long long)NN * 32 + 255) / 256);
    fin2_kernel<<<blocks, 256, 0, stream>>>(agg2, deg, hcat, b2, out);
  }
}